// TokenMoEFFN_9320079033272
// MI455X (gfx1250) — compile-verified
//
#include <hip/hip_runtime.h>
#include <hip/hip_bf16.h>
#include <math.h>

typedef __attribute__((ext_vector_type(2))) float v2f;
typedef __attribute__((ext_vector_type(4))) float v4f;
typedef __attribute__((ext_vector_type(8))) float v8f;

#define DDIM 512
#define HDIM 2048
#define NEXP 8
#define TILE_M 64
#define HCHUNK 64
#define XST (DDIM + 4)     // LDS row stride for x tile (floats)
#define HST (HCHUNK + 4)   // LDS row stride for h chunk (floats)
#define WST_FLOATS (DDIM * HCHUNK)  // 32768 floats = 128KB, fits W1 or W2 chunk

// ---------------------------------------------------------------------------
// Gate: LayerNorm + Linear(D,E) + softmax.  One wave (32 lanes) per token.
// ---------------------------------------------------------------------------
__global__ __launch_bounds__(256) void moe_gate_kernel(
    const float* __restrict__ x, const float* __restrict__ gamma,
    const float* __restrict__ beta, const float* __restrict__ gW,
    const float* __restrict__ gb, float* __restrict__ gates) {
  const int lane = threadIdx.x & 31;
  const int wid = threadIdx.x >> 5;
  const int t = blockIdx.x * 8 + wid;

  const float* xr = x + (size_t)t * DDIM;
  float xv[16];
  float s = 0.f, ss = 0.f;
#pragma unroll
  for (int i = 0; i < 16; ++i) {
    float v = xr[lane + 32 * i];
    xv[i] = v;
    s += v;
    ss += v * v;
  }
#pragma unroll
  for (int m = 16; m >= 1; m >>= 1) {
    s += __shfl_xor(s, m, 32);
    ss += __shfl_xor(ss, m, 32);
  }
  const float mu = s * (1.0f / DDIM);
  const float var = ss * (1.0f / DDIM) - mu * mu;
  const float rs = rsqrtf(var + 1e-5f);

  float lg[NEXP];
#pragma unroll
  for (int e = 0; e < NEXP; ++e) lg[e] = 0.f;
#pragma unroll
  for (int i = 0; i < 16; ++i) {
    const int d = lane + 32 * i;
    const float xn = (xv[i] - mu) * rs * gamma[d] + beta[d];
#pragma unroll
    for (int e = 0; e < NEXP; ++e) lg[e] += xn * gW[d * NEXP + e];
  }
#pragma unroll
  for (int e = 0; e < NEXP; ++e) {
#pragma unroll
    for (int m = 16; m >= 1; m >>= 1) lg[e] += __shfl_xor(lg[e], m, 32);
    lg[e] += gb[e];  // TEMP == 1.0
  }
  float mx = lg[0];
#pragma unroll
  for (int e = 1; e < NEXP; ++e) mx = fmaxf(mx, lg[e]);
  float sum = 0.f;
#pragma unroll
  for (int e = 0; e < NEXP; ++e) {
    lg[e] = expf(lg[e] - mx);
    sum += lg[e];
  }
  const float inv = 1.f / sum;
#pragma unroll
  for (int e = 0; e < NEXP; ++e)
    if (lane == e) gates[(size_t)t * NEXP + e] = lg[e] * inv;
}

// ---------------------------------------------------------------------------
// Fused expert FFN + gate-weighted combine, all V_WMMA_F32_16X16X4_F32.
// Per 64-token tile (256 threads, 8 waves), loop experts x H-chunks(64):
//   stage W1 chunk in LDS (k-interleaved) -> GEMM1 -> h=leaky(.+b1) scaled by
//   gate weight -> LDS; stage W2 chunk (same buffer) -> GEMM2 accumulates the
//   final output tile in 16 persistent v8f accumulators per wave.
// Weight stage layout wst[k>>2][col][k&3] makes every B fragment one
// conflict-free ds_load_b64 (lanes 0-15: k,k+1; lanes 16-31: k+2,k+3).
// ---------------------------------------------------------------------------
__global__ __launch_bounds__(256) void moe_expert_kernel(
    const float* __restrict__ x, const float* __restrict__ gates,
    const float* __restrict__ W1, const float* __restrict__ b1,
    const float* __restrict__ W2, const float* __restrict__ b2,
    float* __restrict__ out) {
  extern __shared__ float smem[];
  float* xs = smem;                         // [TILE_M][XST]      132 KB
  float* wst = xs + TILE_M * XST;           // [WST_FLOATS]       128 KB
  float* hs = wst + WST_FLOATS;             // [TILE_M][HST]     17.4 KB
  float* wsg = hs + TILE_M * HST;           // [TILE_M][NEXP]       2 KB
  float* bs2 = wsg + TILE_M * NEXP;         // [NEXP][DDIM]        16 KB

  const int tid = threadIdx.x;
  const int lane = tid & 31;
  const int wid = tid >> 5;
  const int t0 = blockIdx.x * TILE_M;

  // ---- stage x tile, gate weights, b2 into LDS ----
  for (int idx = tid; idx < TILE_M * (DDIM / 4); idx += 256) {
    const int r = idx / (DDIM / 4);
    const int c4 = idx % (DDIM / 4);
    v4f v = ((const v4f*)(x + (size_t)(t0 + r) * DDIM))[c4];
    *(v4f*)&xs[r * XST + c4 * 4] = v;
  }
  for (int idx = tid; idx < TILE_M * NEXP; idx += 256)
    wsg[idx] = gates[(size_t)t0 * NEXP + idx];
  for (int idx = tid; idx < NEXP * DDIM; idx += 256) bs2[idx] = b2[idx];

  // ---- persistent output accumulators: wave = rows [wm*16,+16),
  //      cols [wn*256,+256) as 16 WMMA tiles ----
  const int wm = wid >> 1;    // 0..3
  const int wn = wid & 1;     // 0..1
  v8f oacc[16];
  {
    v8f z = {};
#pragma unroll
    for (int tt = 0; tt < 16; ++tt) oacc[tt] = z;
  }

  // GEMM1 tile assignment: h chunk is 64x64 = 4x4 tiles of 16x16.
  const int tcol = wid & 3;          // column tile
  const int trow0 = (wid >> 2) * 2;  // two row tiles per wave

  const int am = lane & 15;          // A: M row within tile / B,C: N col
  const int koff = (lane >> 4) * 2;  // K sub-offset per f32 WMMA layout
  const int radd = (lane >> 4) << 3; // C layout: rows j (lanes<16) / j+8

  __syncthreads();

  for (int e = 0; e < NEXP; ++e) {
    const float* __restrict__ W1e = W1 + (size_t)e * DDIM * HDIM;
    const float* __restrict__ W2e = W2 + (size_t)e * HDIM * DDIM;

    // gate weights for this wave's GEMM1 epilogue rows, hoisted to registers
    float gw[16];
#pragma unroll
    for (int j = 0; j < 8; ++j) {
      gw[j] = wsg[(trow0 * 16 + j + radd) * NEXP + e];
      gw[j + 8] = wsg[((trow0 + 1) * 16 + j + radd) * NEXP + e];
    }

    for (int hc0 = 0; hc0 < HDIM; hc0 += HCHUNK) {
      __syncthreads();  // previous GEMM2 done with wst/hs

      // ---- stage W1[:, hc0:+64] into wst[k>>2][c][k&3] ----
      for (int idx = tid; idx < DDIM * (HCHUNK / 4); idx += 256) {
        const int k = idx / (HCHUNK / 4);
        const int c4 = (idx % (HCHUNK / 4)) * 4;
        v4f v = *(const v4f*)&W1e[(size_t)k * HDIM + hc0 + c4];
        float* dst = &wst[(k >> 2) * (HCHUNK * 4) + (k & 3)];
        dst[(c4 + 0) * 4] = v.x;
        dst[(c4 + 1) * 4] = v.y;
        dst[(c4 + 2) * 4] = v.z;
        dst[(c4 + 3) * 4] = v.w;
      }
      __syncthreads();

      // ---------------- GEMM1: 2 tiles per wave, K = 512 ----------------
      {
        v8f acc0 = {}, acc1 = {};
        const float* xr0 = &xs[(trow0 * 16 + am) * XST + koff];
        const float* xr1 = &xs[((trow0 + 1) * 16 + am) * XST + koff];
        const float* bp = &wst[(tcol * 16 + am) * 4 + koff];
#pragma unroll 4
        for (int k = 0; k < DDIM; k += 4) {
          v2f b = *(const v2f*)&bp[(k >> 2) * (HCHUNK * 4)];
          v2f a0 = *(const v2f*)&xr0[k];
          v2f a1 = *(const v2f*)&xr1[k];
          acc0 = __builtin_amdgcn_wmma_f32_16x16x4_f32(false, a0, false, b,
                                                       (short)0, acc0, false,
                                                       false);
          acc1 = __builtin_amdgcn_wmma_f32_16x16x4_f32(false, a1, false, b,
                                                       (short)0, acc1, false,
                                                       false);
        }
        // epilogue: +b1, LeakyReLU(0.01), * gate weight, -> LDS h chunk
        const float bv = b1[(size_t)e * HDIM + hc0 + tcol * 16 + am];
        const int colL = tcol * 16 + am;
#pragma unroll
        for (int j = 0; j < 8; ++j) {
          int r = trow0 * 16 + j + radd;
          float v = acc0[j] + bv;
          v = v > 0.f ? v : 0.01f * v;
          hs[r * HST + colL] = v * gw[j];
          r = (trow0 + 1) * 16 + j + radd;
          v = acc1[j] + bv;
          v = v > 0.f ? v : 0.01f * v;
          hs[r * HST + colL] = v * gw[j + 8];
        }
      }
      __syncthreads();  // GEMM1 done with wst; hs ready

      // ---- stage W2[hc0:+64, :] into wst[k>>2][c][k&3] ----
      for (int idx = tid; idx < HCHUNK * (DDIM / 4); idx += 256) {
        const int k = idx / (DDIM / 4);
        const int c4 = (idx % (DDIM / 4)) * 4;
        v4f v = *(const v4f*)&W2e[(size_t)(hc0 + k) * DDIM + c4];
        float* dst = &wst[(k >> 2) * (DDIM * 4) + (k & 3)];
        dst[(c4 + 0) * 4] = v.x;
        dst[(c4 + 1) * 4] = v.y;
        dst[(c4 + 2) * 4] = v.z;
        dst[(c4 + 3) * 4] = v.w;
      }
      __syncthreads();

      // -------- GEMM2: oacc += h(64x64) @ W2chunk[:, wn*256 : +256) --------
      {
        const float* hr = &hs[(wm * 16 + am) * HST + koff];
        const float* bp = &wst[(wn * 256 + am) * 4 + koff];
        for (int k = 0; k < HCHUNK; k += 4) {
          v2f a = *(const v2f*)&hr[k];
          const float* bk = &bp[(k >> 2) * (DDIM * 4)];
#pragma unroll
          for (int tt = 0; tt < 16; ++tt) {
            v2f b = *(const v2f*)&bk[tt * 64];
            oacc[tt] = __builtin_amdgcn_wmma_f32_16x16x4_f32(
                false, a, false, b, (short)0, oacc[tt], false, false);
          }
        }
      }
    }
  }

  // ---- epilogue: add gate-weighted b2 and store out tile ----
#pragma unroll
  for (int tt = 0; tt < 16; ++tt) {
    const int c = wn * 256 + tt * 16 + am;
#pragma unroll
    for (int j = 0; j < 8; ++j) {
      const int r = wm * 16 + j + radd;
      float v = oacc[tt][j];
#pragma unroll
      for (int e = 0; e < NEXP; ++e) v += wsg[r * NEXP + e] * bs2[e * DDIM + c];
      out[(size_t)(t0 + r) * DDIM + c] = v;
    }
  }
}

// ---------------------------------------------------------------------------
extern "C" void kernel_launch(void* const* d_in, const int* in_sizes, int n_in,
                              void* d_out, int out_size, void* d_ws,
                              size_t ws_size, hipStream_t stream) {
  (void)in_sizes; (void)n_in; (void)out_size; (void)ws_size;
  const float* x = (const float*)d_in[0];
  const float* gamma = (const float*)d_in[1];
  const float* beta = (const float*)d_in[2];
  const float* gW = (const float*)d_in[3];
  const float* gb = (const float*)d_in[4];
  const float* W1 = (const float*)d_in[5];
  const float* b1 = (const float*)d_in[6];
  const float* W2 = (const float*)d_in[7];
  const float* b2 = (const float*)d_in[8];
  float* out = (float*)d_out;
  float* gates = (float*)d_ws;  // T * NEXP floats

  const int T = 4 * 4096;
  moe_gate_kernel<<<T / 8, 256, 0, stream>>>(x, gamma, beta, gW, gb, gates);

  const size_t lds_bytes =
      (size_t)(TILE_M * XST + WST_FLOATS + TILE_M * HST + TILE_M * NEXP +
               NEXP * DDIM) *
      sizeof(float);
  moe_expert_kernel<<<T / TILE_M, 256, lds_bytes, stream>>>(x, gates, W1, b1,
                                                            W2, b2, out);
}